// SplitPool_46961172415286
// MI455X (gfx1250) — compile-verified
//
#include <hip/hip_runtime.h>

// SplitPool for MI455X (gfx1250, wave32):
//   mean over 600-row chunks of 128-dim fp32 vectors, done with exact-fp32
//   V_WMMA_F32_16X16X4_F32 (A = ones => D[m][n] = sum_k B[k][n]).
// Memory-bound: ~1.23 GB read @ 23.3 TB/s => ~53 us floor.

typedef __attribute__((ext_vector_type(2))) float v2f;
typedef __attribute__((ext_vector_type(8))) float v8f;

#define BATCH      8
#define NPEAKS     500
#define CHUNK      600
#define DIM        128
#define PER_SAMPLE (NPEAKS + 1)   // 501 chunks per sample; last one dropped

__global__ __launch_bounds__(256) void splitpool_wmma_f32(
    const float* __restrict__ x,        // (BATCH * PER_SAMPLE * CHUNK, DIM) flat
    const int*   __restrict__ n_peaks,  // (BATCH,)
    float*       __restrict__ out)      // (BATCH, NPEAKS, DIM)
{
    const int blk = blockIdx.x;                 // = b * NPEAKS + p
    const int b   = blk / NPEAKS;
    const int p   = blk - b * NPEAKS;

    const int wave  = threadIdx.x >> 5;         // 0..7  -> dim group of 16
    const int lane  = threadIdx.x & 31;
    const int n     = lane & 15;                // N column within dim group
    const int khalf = lane >> 4;                // 0: K=0,1   1: K=2,3

    // Flat row index of this chunk's first row in the (batch*length, DIM) view.
    const size_t baseRow = (size_t)(b * PER_SAMPLE + p) * CHUNK;
    // Per-lane base pointer: row (baseRow + 2*khalf), column wave*16 + n.
    const float* px = x + baseRow * DIM + (size_t)khalf * 2 * DIM
                        + (size_t)wave * 16 + n;

    const v2f a = {1.0f, 1.0f};                 // A = ones -> column sums
    v8f c0 = {};
    v8f c1 = {};

    // 150 WMMAs of K=4 rows each; 2 accumulators to break the RAW chain.
    #pragma unroll 4
    for (int i = 0; i < CHUNK / 4; i += 2) {
        const float* p0 = px + (size_t)i * 4 * DIM;   // rows 4i + {0,1}/{2,3}
        const float* p1 = p0 + 4 * DIM;               // rows 4i+4 ...

        // Streaming prefetch ~24 KB (48 rows) ahead, once per 32 rows.
        if ((i & 7) == 0)
            __builtin_prefetch(p0 + 48 * DIM, 0, 1);

        v2f b0, b1;
        b0.x = p0[0];
        b0.y = p0[DIM];
        b1.x = p1[0];
        b1.y = p1[DIM];

        c0 = __builtin_amdgcn_wmma_f32_16x16x4_f32(
                 false, a, false, b0, (short)0, c0, false, false);
        c1 = __builtin_amdgcn_wmma_f32_16x16x4_f32(
                 false, a, false, b1, (short)0, c1, false, false);
    }

    v8f cs = c0 + c1;   // element-wise merge of the two D accumulators

    // D layout: VGPR0 (cs[0]), lanes 0-15 hold row M=0, N=lane. All M rows are
    // identical (A=ones), so lanes 0-15 of each wave carry the 16 dim sums.
    const int   np    = n_peaks[b];
    const float scale = (p < np) ? (1.0f / (float)CHUNK) : 0.0f;  // mask => 0

    if (lane < 16) {
        out[(size_t)blk * DIM + wave * 16 + lane] = cs[0] * scale;
    }
}

extern "C" void kernel_launch(void* const* d_in, const int* in_sizes, int n_in,
                              void* d_out, int out_size, void* d_ws, size_t ws_size,
                              hipStream_t stream)
{
    // setup_inputs order: x (f32), chunk_size (scalar), n_peaks (i32[8]), max_n_peaks (scalar)
    const float* x       = (const float*)d_in[0];
    const int*   n_peaks = (const int*)d_in[2];
    float*       out     = (float*)d_out;

    dim3 grid(BATCH * NPEAKS);   // one workgroup per kept chunk
    dim3 block(256);             // 8 wave32 waves: one per 16-dim group
    splitpool_wmma_f32<<<grid, block, 0, stream>>>(x, n_peaks, out);
}